// TAG_27865747817098
// MI455X (gfx1250) — compile-verified
//
#include <hip/hip_runtime.h>
#include <math.h>

typedef float v2f __attribute__((ext_vector_type(2)));
typedef float v8f __attribute__((ext_vector_type(8)));

#define FDIM 31
#define FPAD 32
#define KTOT 93
#define KPAD 96
#define GRAPHS 6400
#define SCAN_T 1024

__device__ __forceinline__ unsigned ordkey(float f) {
  int i = __float_as_int(f);
  return (i < 0) ? ~(unsigned)i : ((unsigned)i | 0x80000000u);
}
__device__ __forceinline__ float ordval(unsigned u) {
  int i = (u & 0x80000000u) ? (int)(u & 0x7fffffffu) : ~(int)u;
  return __int_as_float(i);
}

__global__ void k_zero(float* p, int n) {
  int stride = gridDim.x * blockDim.x;
  for (int i = blockIdx.x * blockDim.x + threadIdx.x; i < n; i += stride) p[i] = 0.0f;
}

// in-degree count (int)
__global__ void k_count(const int* __restrict__ dst, int* __restrict__ ideg, int E) {
  int i = blockIdx.x * blockDim.x + threadIdx.x;
  if (i < E) atomicAdd(&ideg[dst[i]], 1);
}

// norm = rsqrt(clamp(deg,1))
__global__ void k_norm(const int* __restrict__ ideg, float* __restrict__ norm, int N) {
  int i = blockIdx.x * blockDim.x + threadIdx.x;
  if (i < N) norm[i] = rsqrtf(fmaxf((float)ideg[i], 1.0f));
}

// exclusive prefix sum of ideg -> rowstart[0..N], single workgroup hierarchical scan
__global__ void k_scan(const int* __restrict__ ideg, int* __restrict__ rowstart, int N) {
  __shared__ int part[SCAN_T];
  int t = threadIdx.x;
  int chunk = (N + SCAN_T - 1) / SCAN_T;
  int beg = t * chunk;
  int end = beg + chunk; if (end > N) end = N;
  int s = 0;
  for (int i = beg; i < end; ++i) s += ideg[i];
  part[t] = s;
  __syncthreads();
  if (t == 0) {
    int run = 0;
    for (int i = 0; i < SCAN_T; ++i) { int v = part[i]; part[i] = run; run += v; }
    rowstart[N] = run;   // == E
  }
  __syncthreads();
  int run = part[t];
  for (int i = beg; i < end; ++i) { rowstart[i] = run; run += ideg[i]; }
}

// counting-sort edges into CSR-by-dst; precompute fused coefficient norm[s]*norm[d]
__global__ void k_fill(const int* __restrict__ src, const int* __restrict__ dst,
                       const float* __restrict__ norm, const int* __restrict__ rowstart,
                       int* __restrict__ cursor, int* __restrict__ csr_src,
                       float* __restrict__ csr_coef, int E) {
  int i = blockIdx.x * blockDim.x + threadIdx.x;
  if (i >= E) return;
  int s = src[i], d = dst[i];
  int idx = rowstart[d] + atomicAdd(&cursor[d], 1);
  csr_src[idx] = s;
  csr_coef[idx] = norm[s] * norm[d];
}

// pack x [N,31] -> padded [N,32], pad col = 0
__global__ void k_pack(const float* __restrict__ x, float* __restrict__ h, int N) {
  int wid = (blockIdx.x * blockDim.x + threadIdx.x) >> 5;
  int lane = threadIdx.x & 31;
  if (wid < N)
    h[(size_t)wid * FPAD + lane] = (lane < FDIM) ? x[(size_t)wid * FDIM + lane] : 0.0f;
}

// W [L,93,31] -> WT [L][ncol=32][krow=96], zero padded rows/cols.
__global__ void k_prepw(const float* __restrict__ Ws, float* __restrict__ WT, int L) {
  int t = blockIdx.x * blockDim.x + threadIdx.x;
  int total = L * FPAD * KPAD;
  if (t >= total) return;
  int l = t / (FPAD * KPAD);
  int r = t % (FPAD * KPAD);
  int ncol = r / KPAD;
  int kr = r % KPAD;
  int seg = kr >> 5, c = kr & 31;
  float v = 0.0f;
  if (ncol < FDIM && c < FDIM)
    v = Ws[((size_t)l * KTOT + seg * FDIM + c) * FDIM + ncol];
  WT[t] = v;
}

// pull-based SPMM hop: out[v] = sum_{in-edges} coef * in[src]; wave/node, lane/feature.
// No atomics, single coalesced 128B store per node; lane 31 accumulates pad zeros.
__global__ void k_pull(const float* __restrict__ in, float* __restrict__ out,
                       const int* __restrict__ csr_src, const float* __restrict__ csr_coef,
                       const int* __restrict__ rowstart, int N) {
  int lane = threadIdx.x & 31;
  int wid = (blockIdx.x * blockDim.x + threadIdx.x) >> 5;
  if (wid >= N) return;
  int beg = rowstart[wid], end = rowstart[wid + 1];
  float acc = 0.0f;
  for (int j = beg; j < end; ++j) {
    float c = csr_coef[j];          // broadcast
    int s = csr_src[j];             // broadcast
    acc = fmaf(c, in[(size_t)s * FPAD + lane], acc);   // coalesced 128B row gather
  }
  out[(size_t)wid * FPAD + lane] = acc;
}

// out = relu([h|y1|y2] @ W + b) via V_WMMA_F32_16X16X4_F32.
// One wave per 16-node tile; K=96 padded, 24 k-steps; 2 N-tiles (cols 0-15,16-30).
// outh aliases h safely: each wave reads/writes only its own rows, stores after loads.
__global__ void k_gemm(const float* h, const float* f1, const float* f2,
                       const float* __restrict__ WT, const float* __restrict__ bias,
                       float* outh, int N) {
  int wave = threadIdx.x >> 5;
  int lane = threadIdx.x & 31;
  int tile = blockIdx.x * (blockDim.x >> 5) + wave;
  int ntiles = (N + 15) >> 4;
  if (tile >= ntiles) return;
  int row = lane & 15;
  int hi = lane >> 4;
  int hi2 = hi << 1;
  int node = tile * 16 + row;
  if (node >= N) node = N - 1;
  int nc0 = lane & 15;
  int nc1 = nc0 + 16;
  const float* segp[3] = { h + (size_t)node * FPAD,
                           f1 + (size_t)node * FPAD,
                           f2 + (size_t)node * FPAD };
  v8f acc0 = {0.f, 0.f, 0.f, 0.f, 0.f, 0.f, 0.f, 0.f};
  v8f acc1 = {0.f, 0.f, 0.f, 0.f, 0.f, 0.f, 0.f, 0.f};
#pragma unroll
  for (int kk = 0; kk < 24; ++kk) {
    const float* sp = segp[kk >> 3];             // 8 k-steps per 32-col segment
    int col = ((kk & 7) << 2) + hi2;             // A: half-wave K split
    v2f a = *(const v2f*)(sp + col);
    int kw = (kk << 2) + hi2;                    // B mirrors A's K split
    v2f b0 = *(const v2f*)(WT + (size_t)nc0 * KPAD + kw);
    v2f b1 = *(const v2f*)(WT + (size_t)nc1 * KPAD + kw);
    acc0 = __builtin_amdgcn_wmma_f32_16x16x4_f32(false, a, false, b0, (short)0, acc0, false, false);
    acc1 = __builtin_amdgcn_wmma_f32_16x16x4_f32(false, a, false, b1, (short)0, acc1, false, false);
  }
  float bia0 = bias[nc0];
  float bia1 = (nc1 < FDIM) ? bias[nc1] : 0.0f;
  int mbase = tile * 16 + (hi << 3);             // D: VGPR v holds row v (+8 for hi half)
#pragma unroll
  for (int v = 0; v < 8; ++v) {
    int nd = mbase + v;
    if (nd < N) {
      outh[(size_t)nd * FPAD + nc0] = fmaxf(acc0[v] + bia0, 0.0f);
      if (nc1 < FDIM) outh[(size_t)nd * FPAD + nc1] = fmaxf(acc1[v] + bia1, 0.0f);
    }
  }
}

__global__ void k_gate(const float* __restrict__ h, const float* __restrict__ wg,
                       const float* __restrict__ bg, float* __restrict__ gate,
                       unsigned* __restrict__ gmax, const int* __restrict__ gids, int N) {
  int wid = (blockIdx.x * blockDim.x + threadIdx.x) >> 5;
  int lane = threadIdx.x & 31;
  if (wid >= N) return;
  float v = (lane < FDIM) ? h[(size_t)wid * FPAD + lane] * wg[lane] : 0.0f;
  for (int o = 16; o > 0; o >>= 1) v += __shfl_down(v, o);
  if (lane == 0) {
    float g = v + bg[0];
    gate[wid] = g;
    atomicMax(&gmax[gids[wid]], ordkey(g));
  }
}

__global__ void k_exp(float* __restrict__ gate, const unsigned* __restrict__ gmax,
                      float* __restrict__ den, const int* __restrict__ gids, int N) {
  int i = blockIdx.x * blockDim.x + threadIdx.x;
  if (i >= N) return;
  int g = gids[i];
  float e = expf(gate[i] - ordval(gmax[g]));
  gate[i] = e;
  atomicAdd(&den[g], e);
}

__global__ void k_rep(const float* __restrict__ h, const float* __restrict__ ebuf,
                      const float* __restrict__ den, const int* __restrict__ gids,
                      float* __restrict__ rep, int N) {
  int wid = (blockIdx.x * blockDim.x + threadIdx.x) >> 5;
  int lane = threadIdx.x & 31;
  if (wid >= N) return;
  int g = gids[wid];
  float alpha = ebuf[wid] / den[g];
  if (lane < FDIM)
    atomicAdd(&rep[(size_t)g * FDIM + lane], alpha * h[(size_t)wid * FPAD + lane]);
}

__global__ void k_scatter(const float* __restrict__ rep, const int* __restrict__ num,
                          float* __restrict__ out, int G, int B, int ostride) {
  int t = blockIdx.x * blockDim.x + threadIdx.x;
  if (t >= G * FDIM) return;
  int g = t / FDIM, f = t % FDIM;
  int off = 0, s = 0;
  for (; s < B - 1; ++s) {
    int nn = num[s];
    if (g < off + nn) break;
    off += nn;
  }
  out[(size_t)s * ostride + (size_t)(g - off) * FDIM + f] = rep[t];
}

extern "C" void kernel_launch(void* const* d_in, const int* in_sizes, int n_in,
                              void* d_out, int out_size, void* d_ws, size_t ws_size,
                              hipStream_t stream) {
  const float* x   = (const float*)d_in[0];
  const float* Ws  = (const float*)d_in[1];
  const float* bs  = (const float*)d_in[2];
  const float* wg  = (const float*)d_in[3];
  const float* bg  = (const float*)d_in[4];
  const int*   src = (const int*)d_in[5];
  const int*   dst = (const int*)d_in[6];
  const int*   gid = (const int*)d_in[7];
  const int*   num = (const int*)d_in[8];

  const int N = in_sizes[7];
  const int E = in_sizes[5];
  const int B = in_sizes[8];
  const int L = in_sizes[1] / (KTOT * FDIM);
  const int G = GRAPHS;
  const int ostride = out_size / B;

  float* w     = (float*)d_ws;
  float* norm  = w;                                // N
  float* hbuf  = norm + N;                         // N*32
  float* y1    = hbuf + (size_t)N * FPAD;          // N*32
  float* y2    = y1 + (size_t)N * FPAD;            // N*32
  float* WT    = y2 + (size_t)N * FPAD;            // L*32*96
  float* gate  = WT + (size_t)L * FPAD * KPAD;     // N
  unsigned* gmax = (unsigned*)(gate + N);          // G
  float* den   = (float*)(gmax + G);               // G
  float* rep   = den + G;                          // G*31
  int*   ideg  = (int*)(rep + (size_t)G * FDIM);   // N
  int*   rowstart = ideg + N;                      // N+1
  int*   cursor   = rowstart + N + 1;              // N
  int*   csr_src  = cursor + N;                    // E
  float* csr_coef = (float*)(csr_src + E);         // E

  const int T = 256;

  k_zero<<<2048, T, 0, stream>>>((float*)ideg, N);
  k_zero<<<2048, T, 0, stream>>>((float*)cursor, N);
  k_zero<<<64, T, 0, stream>>>((float*)gmax, G);
  k_zero<<<64, T, 0, stream>>>(den, G);
  k_zero<<<512, T, 0, stream>>>(rep, G * FDIM);
  k_zero<<<1024, T, 0, stream>>>((float*)d_out, out_size);

  // CSR-by-dst build (once per call)
  k_count<<<(E + T - 1) / T, T, 0, stream>>>(dst, ideg, E);
  k_norm<<<(N + T - 1) / T, T, 0, stream>>>(ideg, norm, N);
  k_scan<<<1, SCAN_T, 0, stream>>>(ideg, rowstart, N);
  k_fill<<<(E + T - 1) / T, T, 0, stream>>>(src, dst, norm, rowstart, cursor,
                                            csr_src, csr_coef, E);

  k_pack<<<(N * 32 + T - 1) / T, T, 0, stream>>>(x, hbuf, N);
  k_prepw<<<(L * FPAD * KPAD + T - 1) / T, T, 0, stream>>>(Ws, WT, L);

  int waveBlocks = (N * 32 + T - 1) / T;           // one wave per node
  int gemmBlocks = (((N + 15) / 16) + 7) / 8;      // 8 tiles (waves) per block
  for (int l = 0; l < L; ++l) {
    k_pull<<<waveBlocks, T, 0, stream>>>(hbuf, y1, csr_src, csr_coef, rowstart, N);
    k_pull<<<waveBlocks, T, 0, stream>>>(y1, y2, csr_src, csr_coef, rowstart, N);
    k_gemm<<<gemmBlocks, T, 0, stream>>>(hbuf, y1, y2,
                                         WT + (size_t)l * FPAD * KPAD,
                                         bs + (size_t)l * FDIM, hbuf, N);
  }

  k_gate<<<waveBlocks, T, 0, stream>>>(hbuf, wg, bg, gate, gmax, gid, N);
  k_exp<<<(N + T - 1) / T, T, 0, stream>>>(gate, gmax, den, gid, N);
  k_rep<<<waveBlocks, T, 0, stream>>>(hbuf, gate, den, gid, rep, N);
  k_scatter<<<(G * FDIM + T - 1) / T, T, 0, stream>>>(rep, num, (float*)d_out, G, B, ostride);
}